// KernelDensity_17239998726765
// MI455X (gfx1250) — compile-verified
//
#include <hip/hip_runtime.h>
#include <hip/hip_bf16.h>

typedef __attribute__((ext_vector_type(2))) float v2f;
typedef __attribute__((ext_vector_type(8))) float v8f;

#define KDE_Q 4096
#define KDE_N 65536
#define KDE_D 32
#define KDE_LOG2E 1.4426950408889634f

// ---------------------------------------------------------------------------
// c_j = log2(e) * ( -0.5*||t_j||^2 + ln(w_j) )   (folded into WMMA C-init)
// ---------------------------------------------------------------------------
__global__ __launch_bounds__(256) void kde_pre(const float* __restrict__ XT,
                                               const float* __restrict__ w,
                                               float* __restrict__ c) {
  int j = blockIdx.x * 256 + threadIdx.x;
  const float* row = XT + (size_t)j * KDE_D;
  float t2 = 0.f;
#pragma unroll
  for (int d = 0; d < KDE_D; ++d) { float v = row[d]; t2 += v * v; }
  c[j] = KDE_LOG2E * (-0.5f * t2 + __logf(w[j]));
}

// ---------------------------------------------------------------------------
// Deterministic single-block reduction: W = sum_j w_j
// ---------------------------------------------------------------------------
__global__ __launch_bounds__(1024) void kde_wsum(const float* __restrict__ w,
                                                 float* __restrict__ W) {
  __shared__ float lds[32];
  float s = 0.f;
  for (int i = threadIdx.x; i < KDE_N; i += 1024) s += w[i];
#pragma unroll
  for (int off = 16; off >= 1; off >>= 1) s += __shfl_xor(s, off, 32);
  if ((threadIdx.x & 31) == 0) lds[threadIdx.x >> 5] = s;
  __syncthreads();
  if (threadIdx.x < 32) {
    float t = lds[threadIdx.x];
#pragma unroll
    for (int off = 16; off >= 1; off >>= 1) t += __shfl_xor(t, off, 32);
    if (threadIdx.x == 0) W[0] = t;
  }
}

// ---------------------------------------------------------------------------
// Main: one 16-query tile per block, full N-sweep split over 16 waves.
// Two WMMA accumulator chains (N-tiles n0, n0+16) + double-buffered B
// prefetch: next iteration's 16 b64 loads are in flight during the entire
// current 16-WMMA + 16-exp block, hiding L2 latency.
// S[i] = sum_j exp2( log2e*(x_i . t_j) + c_j ) = sum_j w_j exp(x.t - 0.5||t||^2)
// ---------------------------------------------------------------------------
__global__ __launch_bounds__(512) void kde_main(const float* __restrict__ X,
                                                const float* __restrict__ XT,
                                                const float* __restrict__ c,
                                                float* __restrict__ S) {
  __shared__ float red[16][16];
  const int lane = threadIdx.x & 31;
  const int wave = threadIdx.x >> 5;   // 0..15
  const int r    = lane & 15;          // row / column slot within 16-lane half
  const int half = lane >> 4;          // 0 or 1
  const int koff = half * 2;           // K-element offset within group of 4
  const int q0   = blockIdx.x * 16;

  // A tile (16 queries x 32 dims), pre-scaled by log2(e).
  // ISA 16x4 f32 A layout: lanes 0-15 hold K={0,1}, lanes 16-31 hold K={2,3}.
  v2f A[8];
  const float* arow = X + (size_t)(q0 + r) * KDE_D + koff;
#pragma unroll
  for (int kk = 0; kk < 8; ++kk) {
    v2f a = *(const v2f*)(arow + 4 * kk);
    A[kk] = a * KDE_LOG2E;
  }

  v8f sum0 = {};
  v8f sum1 = {};

  const int n_per_wave = KDE_N / 16;       // 4096 train points per wave
  const int n_begin = wave * n_per_wave;
  const int n_end   = n_begin + n_per_wave;

  // Prologue: load first 2-tile B block + c values.
  v2f B0[8], B1[8];
  float c0, c1;
  {
    const float* b0 = XT + (size_t)(n_begin + r) * KDE_D + koff;
    const float* b1 = XT + (size_t)(n_begin + 16 + r) * KDE_D + koff;
#pragma unroll
    for (int kk = 0; kk < 8; ++kk) {
      B0[kk] = *(const v2f*)(b0 + 4 * kk);
      B1[kk] = *(const v2f*)(b1 + 4 * kk);
    }
    c0 = c[n_begin + r];
    c1 = c[n_begin + 16 + r];
  }

  for (int n0 = n_begin; n0 < n_end; n0 += 32) {
    // Prefetch next iteration (wrap to n_begin on last iter: redundant but
    // keeps control flow uniform and addresses in range).
    const int nn = (n0 + 32 < n_end) ? (n0 + 32) : n_begin;
    v2f NB0[8], NB1[8];
    {
      const float* b0 = XT + (size_t)(nn + r) * KDE_D + koff;
      const float* b1 = XT + (size_t)(nn + 16 + r) * KDE_D + koff;
#pragma unroll
      for (int kk = 0; kk < 8; ++kk) {
        NB0[kk] = *(const v2f*)(b0 + 4 * kk);
        NB1[kk] = *(const v2f*)(b1 + 4 * kk);
      }
    }
    float nc0 = c[nn + r];
    float nc1 = c[nn + 16 + r];

    // Compute on current block: two independent WMMA chains.
    v8f acc0 = {c0, c0, c0, c0, c0, c0, c0, c0};
    v8f acc1 = {c1, c1, c1, c1, c1, c1, c1, c1};
#pragma unroll
    for (int kk = 0; kk < 8; ++kk) {
      acc0 = __builtin_amdgcn_wmma_f32_16x16x4_f32(
          false, A[kk], false, B0[kk], (short)0, acc0, false, false);
      acc1 = __builtin_amdgcn_wmma_f32_16x16x4_f32(
          false, A[kk], false, B1[kk], (short)0, acc1, false, false);
    }
    // Epilogue: 1 trans + 1 add per element (base-2 exp, scale pre-folded).
#pragma unroll
    for (int i = 0; i < 8; ++i) {
      sum0[i] += __builtin_amdgcn_exp2f(acc0[i]);
      sum1[i] += __builtin_amdgcn_exp2f(acc1[i]);
    }

    // Rotate prefetch buffers.
#pragma unroll
    for (int kk = 0; kk < 8; ++kk) { B0[kk] = NB0[kk]; B1[kk] = NB1[kk]; }
    c0 = nc0;
    c1 = nc1;
  }

  // Reduce the 16 columns held across each 16-lane half.
  // VGPR i of a half-wave corresponds to query row i (+8 for upper half).
#pragma unroll
  for (int i = 0; i < 8; ++i) {
    float s = sum0[i] + sum1[i];
    s += __shfl_xor(s, 1, 32);
    s += __shfl_xor(s, 2, 32);
    s += __shfl_xor(s, 4, 32);
    s += __shfl_xor(s, 8, 32);
    if (r == 0) red[wave][i + half * 8] = s;
  }
  __syncthreads();
  if (threadIdx.x < 16) {
    float acc = 0.f;
#pragma unroll
    for (int wv = 0; wv < 16; ++wv) acc += red[wv][threadIdx.x];
    S[q0 + threadIdx.x] = acc;
  }
}

// ---------------------------------------------------------------------------
// out_i = ln(S_i) - 0.5*||x_i||^2 - 16*ln(2*pi) - ln(W)
// ---------------------------------------------------------------------------
__global__ __launch_bounds__(256) void kde_fin(const float* __restrict__ X,
                                               const float* __restrict__ S,
                                               const float* __restrict__ W,
                                               float* __restrict__ out) {
  int i = blockIdx.x * 256 + threadIdx.x;
  const float* row = X + (size_t)i * KDE_D;
  float q2 = 0.f;
#pragma unroll
  for (int d = 0; d < KDE_D; ++d) { float v = row[d]; q2 += v * v; }
  const float C = 16.0f * 1.8378770664093453f;  // 0.5 * D * ln(2*pi)
  out[i] = __logf(S[i]) - 0.5f * q2 - C - __logf(W[0]);
}

extern "C" void kernel_launch(void* const* d_in, const int* in_sizes, int n_in,
                              void* d_out, int out_size, void* d_ws, size_t ws_size,
                              hipStream_t stream) {
  (void)in_sizes; (void)n_in; (void)out_size; (void)ws_size;
  const float* X  = (const float*)d_in[0];   // [4096, 32]
  const float* XT = (const float*)d_in[1];   // [65536, 32]
  const float* w  = (const float*)d_in[2];   // [65536]
  float* out = (float*)d_out;                // [4096]

  float* S = (float*)d_ws;                   // [4096]
  float* c = S + KDE_Q;                      // [65536]
  float* W = c + KDE_N;                      // [1]

  kde_pre <<<KDE_N / 256, 256, 0, stream>>>(XT, w, c);
  kde_wsum<<<1, 1024, 0, stream>>>(w, W);
  kde_main<<<KDE_Q / 16, 512, 0, stream>>>(X, XT, c, S);
  kde_fin <<<KDE_Q / 256, 256, 0, stream>>>(X, S, W, out);
}